// Proposal_16913581211651
// MI455X (gfx1250) — compile-verified
//
#include <hip/hip_runtime.h>
#include <hip/hip_bf16.h>
#include <stdint.h>

// Proposal generation (softmax score -> box decode -> sort -> greedy NMS).
// CDNA5 (gfx1250) specifics used:
//   * wave32-sized cooperative kernels (1024-thread = 32-wave workgroup)
//   * Tensor Data Mover (tensor_load_to_lds) to async-stream sorted box tiles
//     into LDS, double-buffered, synchronized with s_wait_tensorcnt
//   * all NMS hot state (selected boxes, candidate tiles, valid bitmask) in LDS

#define NMS_THR   0.6f
#define BB        1024        // candidate tile size (boxes) == block threads
#define MAXSEL    1024        // max selections held in LDS (P = 1000 <= 1024)
#define SORT_TILE 2048        // LDS bitonic tile

#if defined(__gfx1250__) && __has_builtin(__builtin_amdgcn_tensor_load_to_lds) && __has_builtin(__builtin_amdgcn_s_wait_tensorcnt)
#define USE_TDM 1
#else
#define USE_TDM 0
#endif

typedef unsigned int tdm_u32x4 __attribute__((ext_vector_type(4)));
typedef int          tdm_i32x8 __attribute__((ext_vector_type(8)));
typedef int          tdm_i32x4 __attribute__((ext_vector_type(4)));

#if USE_TDM
// 1D TDM load: n_dwords 32-bit words from global gsrc -> LDS at lds_ptr.
// D# per CDNA5 ISA ch.8: group0 = {count=1, lds_addr, global_addr[56:0], type=2}
// group1 = {data_size=4B, tensor_dim0=n_dwords, tensor_dim1=1, tile_dim0=n_dwords,
//           tile_dim1=1, tensor_dim0_stride=n_dwords}; groups 2/3 = 0 (<=2D).
// This toolchain (clang-23 / therock-10.0) uses the 6-arg builtin form:
//   (uint32x4 g0, int32x8 g1, int32x4, int32x4, int32x8, i32 cpol)
__device__ __forceinline__ void tdm_issue_1d(const void* gsrc, const void* lds_ptr,
                                             unsigned n_dwords) {
  unsigned long long ga = (unsigned long long)gsrc;
  unsigned lds = (unsigned)(unsigned long long)lds_ptr;   // generic low 32 = LDS byte addr
  tdm_u32x4 g0;
  g0[0] = 1u;                                              // count=1, user mode
  g0[1] = lds;                                             // lds_addr (bytes)
  g0[2] = (unsigned)ga;                                    // global_addr[31:0]
  g0[3] = (unsigned)((ga >> 32) & 0x01FFFFFFull) | (2u << 30); // [56:32] + type=2
  tdm_i32x8 g1;
  g1[0] = (int)(2u << 16);                                 // data_size = 2 (4 bytes)
  g1[1] = (int)((n_dwords & 0xFFFFu) << 16);               // tensor_dim0[15:0]
  g1[2] = (int)((n_dwords >> 16) | (1u << 16));            // tensor_dim0[31:16], tensor_dim1=1
  g1[3] = (int)(n_dwords << 16);                           // tile_dim0 (16-bit field)
  g1[4] = 1;                                               // tile_dim1=1, tile_dim2=0
  g1[5] = (int)n_dwords;                                   // tensor_dim0_stride[31:0]
  g1[6] = 0;                                               // stride hi / dim1_stride
  g1[7] = 0;
  tdm_i32x4 z4; z4[0] = 0; z4[1] = 0; z4[2] = 0; z4[3] = 0;
  tdm_i32x8 z8; z8[0] = 0; z8[1] = 0; z8[2] = 0; z8[3] = 0;
  z8[4] = 0; z8[5] = 0; z8[6] = 0; z8[7] = 0;
  __builtin_amdgcn_tensor_load_to_lds(g0, g1, z4, z4, z8, 0);
}
#endif

// ---------------- phase 1: score + box decode + sort keys ----------------
__global__ void decode_score_kernel(const float* __restrict__ match,
                                    const float* __restrict__ delta,
                                    const float* __restrict__ anc,
                                    float4* __restrict__ boxes,
                                    unsigned long long* __restrict__ keys, int N) {
  int i = blockIdx.x * blockDim.x + threadIdx.x;
  if (i >= N) return;
  float2 m = ((const float2*)match)[i];
  float score = 1.0f / (1.0f + expf(m.x - m.y));   // softmax(match)[:,1]
  float4 a = ((const float4*)anc)[i];
  float4 d = ((const float4*)delta)[i];
  float h = a.z - a.x, w = a.w - a.y;
  float cy = a.x + 0.5f * h + d.x * h;
  float cx = a.y + 0.5f * w + d.y * w;
  h *= expf(d.z);
  w *= expf(d.w);
  float y1 = cy - 0.5f * h, x1 = cx - 0.5f * w;
  boxes[i] = make_float4(y1, x1, y1 + h, x1 + w);
  // monotone key: ascending u64 sort == descending score, stable by index
  unsigned sb = __float_as_uint(score);
  sb = (sb & 0x80000000u) ? ~sb : (sb | 0x80000000u);  // float -> ascending uint
  keys[i] = ((unsigned long long)(~sb) << 32) | (unsigned)i;
}

// ---------------- phase 2: bitonic sort (hybrid LDS/global) ----------------
__device__ __forceinline__ void cmp_xchg(unsigned long long* lk, int i, int l, bool up) {
  unsigned long long a = lk[i], b = lk[l];
  if ((a > b) == up) { lk[i] = b; lk[l] = a; }
}

__global__ __launch_bounds__(1024) void bitonic_local_sort(unsigned long long* __restrict__ keys) {
  __shared__ unsigned long long lk[SORT_TILE];
  int tid = threadIdx.x;
  int base = blockIdx.x * SORT_TILE;
  lk[tid] = keys[base + tid];
  lk[tid + 1024] = keys[base + tid + 1024];
  __syncthreads();
  for (int k = 2; k <= SORT_TILE; k <<= 1) {
    for (int j = k >> 1; j > 0; j >>= 1) {
      int i = ((tid & ~(j - 1)) << 1) | (tid & (j - 1));
      cmp_xchg(lk, i, i + j, ((base + i) & k) == 0);
      __syncthreads();
    }
  }
  keys[base + tid] = lk[tid];
  keys[base + tid + 1024] = lk[tid + 1024];
}

__global__ __launch_bounds__(1024) void bitonic_local_merge(unsigned long long* __restrict__ keys, int k) {
  __shared__ unsigned long long lk[SORT_TILE];
  int tid = threadIdx.x;
  int base = blockIdx.x * SORT_TILE;
  lk[tid] = keys[base + tid];
  lk[tid + 1024] = keys[base + tid + 1024];
  __syncthreads();
  for (int j = SORT_TILE / 2; j > 0; j >>= 1) {
    int i = ((tid & ~(j - 1)) << 1) | (tid & (j - 1));
    cmp_xchg(lk, i, i + j, ((base + i) & k) == 0);
    __syncthreads();
  }
  keys[base + tid] = lk[tid];
  keys[base + tid + 1024] = lk[tid + 1024];
}

__global__ void bitonic_global_pass(unsigned long long* __restrict__ keys, int k, int j) {
  int t = blockIdx.x * blockDim.x + threadIdx.x;
  int i = ((t & ~(j - 1)) << 1) | (t & (j - 1));
  int l = i + j;
  bool up = ((i & k) == 0);
  unsigned long long a = keys[i], b = keys[l];
  if ((a > b) == up) { keys[i] = b; keys[l] = a; }
}

// ---------------- phase 3: gather boxes into sorted order ----------------
__global__ void gather_kernel(const unsigned long long* __restrict__ keys,
                              const float4* __restrict__ boxes,
                              float4* __restrict__ sboxes, int N) {
  int i = blockIdx.x * blockDim.x + threadIdx.x;
  if (i >= N) return;
  if (i + 4096 < N) __builtin_prefetch(keys + i + 4096, 0, 1);  // global_prefetch_b8
  sboxes[i] = boxes[(unsigned)(keys[i] & 0xFFFFFFFFull)];
}

// ---------------- phase 4: greedy NMS (single WG, TDM-streamed tiles) -------
__device__ __forceinline__ float iou_box(float4 a, float aA, float4 b, float bA) {
  float iw = fminf(a.z, b.z) - fmaxf(a.x, b.x);   // y-extent (c0/c2 in reference)
  float ih = fminf(a.w, b.w) - fmaxf(a.y, b.y);   // x-extent (c1/c3)
  float inter = fmaxf(iw, 0.0f) * fmaxf(ih, 0.0f);
  return inter / (aA + bA - inter);
}

__device__ __forceinline__ void tile_issue(const float4* gsrc, float4* lbuf) {
#if USE_TDM
  if (threadIdx.x == 0) tdm_issue_1d(gsrc, lbuf, BB * 4u);   // tensor_load_to_lds
#else
  (void)gsrc; (void)lbuf;
#endif
}

__device__ __forceinline__ void tile_wait(const float4* gsrc, float4* lbuf) {
#if USE_TDM
  if (threadIdx.x == 0) __builtin_amdgcn_s_wait_tensorcnt(0);
  (void)gsrc;
#else
  lbuf[threadIdx.x] = gsrc[threadIdx.x];   // synchronous fallback
#endif
}

__global__ __launch_bounds__(1024) void nms_kernel(const float4* __restrict__ sboxes,
                                                   float* __restrict__ out, int N, int P) {
  __shared__ float4  sel[MAXSEL];
  __shared__ float   selArea[MAXSEL];
  __shared__ float4  cand[2][BB];        // double-buffered TDM destination
  __shared__ unsigned passMask[BB / 32];
  __shared__ int s_selCount, s_next, s_done;

  const int tid = threadIdx.x;
  if (tid == 0) { s_selCount = 0; s_done = 0; }
  if (P > MAXSEL) P = MAXSEL;
  const int numTiles = N / BB;

  tile_issue(sboxes, &cand[0][0]);       // prime pipeline with tile 0
  __syncthreads();

  for (int t = 0; t < numTiles; ++t) {
    const int buf = t & 1;
    tile_wait(sboxes + (size_t)t * BB, &cand[buf][0]);
    __syncthreads();                     // tile t visible to all waves
    if (t + 1 < numTiles)
      tile_issue(sboxes + (size_t)(t + 1) * BB, &cand[buf ^ 1][0]);  // overlap

    // --- phase A: parallel test vs all previously selected boxes (LDS) ---
    float4 my = cand[buf][tid];
    float myArea = (my.z - my.x) * (my.w - my.y);
    int sc0 = s_selCount;
    bool ok = true;
    for (int s = 0; s < sc0; ++s) {
      if (iou_box(my, myArea, sel[s], selArea[s]) > NMS_THR) { ok = false; break; }
    }
    if (tid < BB / 32) passMask[tid] = 0xFFFFFFFFu;
    __syncthreads();
    if (!ok) atomicAnd(&passMask[tid >> 5], ~(1u << (tid & 31)));
    __syncthreads();

    // --- phase B: sequential intra-tile selection via LDS bitmask ---
    for (;;) {
      if (tid == 0) {
        int nxt = -1;
        if (s_selCount < P) {
          for (int w = 0; w < BB / 32; ++w) {
            unsigned m = passMask[w];
            if (m) { nxt = w * 32 + (__ffs(m) - 1); passMask[w] = m & (m - 1); break; }
          }
        }
        s_next = nxt;
        if (nxt >= 0) {
          float4 bx = cand[buf][nxt];
          sel[s_selCount] = bx;
          selArea[s_selCount] = (bx.z - bx.x) * (bx.w - bx.y);
          s_selCount++;
        } else if (s_selCount >= P) {
          s_done = 1;
        }
      }
      __syncthreads();
      int c = s_next;
      if (c < 0) break;
      if (tid > c) {
        unsigned bit = 1u << (tid & 31);
        if (passMask[tid >> 5] & bit) {
          float4 bx = cand[buf][c];
          float bA = (bx.z - bx.x) * (bx.w - bx.y);
          if (iou_box(my, myArea, bx, bA) > NMS_THR)
            atomicAnd(&passMask[tid >> 5], ~bit);
        }
      }
      __syncthreads();
    }
    if (s_done) break;                   // uniform: read after barrier
  }

#if USE_TDM
  if (tid == 0) __builtin_amdgcn_s_wait_tensorcnt(0);   // drain outstanding TDM
#endif
  __syncthreads();

  const int sc = s_selCount;
  for (int k = tid; k < P; k += BB) {
    float4 v = (k < sc) ? sel[k] : make_float4(0.f, 0.f, 0.f, 0.f);
    ((float4*)out)[k] = v;
  }
}

// ---------------- host-side launch ----------------
extern "C" void kernel_launch(void* const* d_in, const int* in_sizes, int n_in,
                              void* d_out, int out_size, void* d_ws, size_t ws_size,
                              hipStream_t stream) {
  const float* match   = (const float*)d_in[0];
  const float* bbox    = (const float*)d_in[1];
  const float* anchors = (const float*)d_in[2];
  // d_in[3] img_shape and d_in[4] num_proposals unused (P from out_size)
  const int N = in_sizes[0] / 2;          // 131072
  const int P = out_size / 4;             // 1000

  // workspace: keys (N*8) | boxes (N*16) | sorted boxes (N*16)  = N*40 bytes
  char* ws = (char*)d_ws;
  unsigned long long* keys = (unsigned long long*)ws;
  float4* boxes  = (float4*)(ws + (size_t)N * 8);
  float4* sboxes = (float4*)(ws + (size_t)N * 8 + (size_t)N * 16);

  decode_score_kernel<<<(N + 255) / 256, 256, 0, stream>>>(match, bbox, anchors, boxes, keys, N);

  // bitonic sort: local 2048-tiles, then global strides >= 2048 + local merges
  bitonic_local_sort<<<N / SORT_TILE, 1024, 0, stream>>>(keys);
  for (int k = SORT_TILE * 2; k <= N; k <<= 1) {
    for (int j = k >> 1; j >= SORT_TILE; j >>= 1)
      bitonic_global_pass<<<(N / 2) / 256, 256, 0, stream>>>(keys, k, j);
    bitonic_local_merge<<<N / SORT_TILE, 1024, 0, stream>>>(keys, k);
  }

  gather_kernel<<<(N + 255) / 256, 256, 0, stream>>>(keys, boxes, sboxes, N);

  nms_kernel<<<1, BB, 0, stream>>>(sboxes, (float*)d_out, N, P);
}